// SolveScheduling_48180943126990
// MI455X (gfx1250) — compile-verified
//
#include <hip/hip_runtime.h>
#include <math.h>

#define NN   24      // horizon
#define NC   23      // N-1 ramp rows per sign
#define TPB  64      // small blocks -> spread 128 blocks over many WGPs
#define LSTR 65      // padded LDS stride (floats), odd -> conflict-free lane reads

#if defined(__gfx1250__) && \
    __has_builtin(__builtin_amdgcn_global_load_async_to_lds_b32) && \
    __has_builtin(__builtin_amdgcn_s_wait_asynccnt)
#define USE_ASYNC_LDS 1
#else
#define USE_ASYNC_LDS 0
#endif

typedef __attribute__((address_space(1))) int* gint_t;   // global src (non-const per builtin)
typedef __attribute__((address_space(3))) int* lint_t;   // LDS dst

__device__ __forceinline__ float frcp(float x) { return __builtin_amdgcn_rcpf(x); }

__global__ __launch_bounds__(TPB)
void sqp_ipm_kernel(const float* __restrict__ gmu,
                    const float* __restrict__ gsig,
                    float* __restrict__ gout, int batch)
{
    __shared__ float smu [NN * LSTR];
    __shared__ float ssig[NN * LSTR];

    const int t    = threadIdx.x;
    const int base = blockIdx.x * TPB;          // first problem of this block

    // ---- stage the block's [TPB x NN] tiles, transposed into LDS.
    // Coalesced global access; on gfx1250 use async global->LDS (ASYNCcnt path).
    #pragma unroll
    for (int j = 0; j < NN; ++j) {
        int g = j * TPB + t;                    // flat element within tile
        int r = g / NN;                         // problem within block
        int c = g - r * NN;                     // horizon index
        int row  = base + r;
        int loff = c * LSTR + r;                // transposed, padded LDS slot
#if USE_ASYNC_LDS
        if (row < batch) {
            __builtin_amdgcn_global_load_async_to_lds_b32(
                (gint_t)(float*)(gmu  + (size_t)row * NN + c), (lint_t)&smu [loff], 0, 0);
            __builtin_amdgcn_global_load_async_to_lds_b32(
                (gint_t)(float*)(gsig + (size_t)row * NN + c), (lint_t)&ssig[loff], 0, 0);
        }
#else
        float mv = 0.f, sv = 1.f;
        if (row < batch) { mv = gmu[(size_t)row*NN + c]; sv = gsig[(size_t)row*NN + c]; }
        smu [loff] = mv;
        ssig[loff] = sv;
#endif
    }
#if USE_ASYNC_LDS
    __builtin_amdgcn_s_wait_asynccnt(0);
#endif
    __syncthreads();

    const int prob = base + t;

    float zz[NN];                // SQP carry, reused as QP iterate
    float qd[NN], p[NN];
    float sA[NC], sB[NC], lA[NC], lB[NC];

    #pragma unroll
    for (int k = 0; k < NN; ++k) zz[k] = smu[k*LSTR + t];

    const float C            = 0.4f;
    const float INV_SQRT2    = 0.70710678118654752f;
    const float INV_SQRT_2PI = 0.39894228040143268f;

    #pragma unroll 1
    for (int it = 0; it < 21; ++it) {           // 20 scan steps + final live step
        // ---- SQP linearization at z = zz (dg, d2g -> qd, p)
        #pragma unroll
        for (int k = 0; k < NN; ++k) {
            float m   = smu [k*LSTR + t];
            float isg = frcp(ssig[k*LSTR + t]);
            float tt  = (zz[k] - m) * isg;
            float cdf = 0.5f * (1.0f + erff(tt * INV_SQRT2));
            float pdf = INV_SQRT_2PI * __expf(-0.5f*tt*tt) * isg;
            float dg  = 50.5f * cdf - 50.0f;
            float d2g = 50.5f * pdf;
            qd[k] = d2g + 1.0f;
            p[k]  = dg - d2g * zz[k] - m;
        }
        // ---- QP init: z = 0 (strictly feasible), s = h, lam = 1
        #pragma unroll
        for (int k = 0; k < NN; ++k) zz[k] = 0.f;
        #pragma unroll
        for (int k = 0; k < NC; ++k) { sA[k]=C; sB[k]=C; lA[k]=1.f; lB[k]=1.f; }

        #pragma unroll 1
        for (int ni = 0; ni < 20; ++ni) {
            // centering target
            float dot = 0.f;
            #pragma unroll
            for (int k = 0; k < NC; ++k) dot += sA[k]*lA[k] + sB[k]*lB[k];
            float muc = 0.1f * dot * (1.0f/46.0f);

            // fused: build tridiagonal M & rhs row-by-row + Thomas forward sweep
            float cp[NC], dp[NN];
            float w_prev = 0.f, u_prev = 0.f, cp_prev = 0.f, dp_prev = 0.f;
            #pragma unroll
            for (int j = 0; j < NN; ++j) {
                float w = 0.f, u = 0.f;
                if (j < NC) {
                    float d  = zz[j] - zz[j+1];
                    float ra =  d + sA[j] - C;          // r_p (upper block)
                    float rb = -d + sB[j] - C;          // r_p (lower block)
                    float ia = frcp(sA[j]);
                    float ib = frcp(sB[j]);
                    float wA = lA[j]*ia, wB = lB[j]*ib;
                    w = wA + wB;                        // tridiag weight
                    u = (muc*ia + wA*ra) - (muc*ib + wB*rb);  // (lam+q) diff
                }
                float diag = qd[j] + w_prev + w;
                float off  = -w_prev;
                float rhs  = -(qd[j]*zz[j] + p[j] + (u - u_prev)); // -(r_d + G^T q)
                float den  = diag - off*cp_prev;
                float inv  = frcp(den);
                float cpj  = -w * inv;
                float dpj  = (rhs - off*dp_prev) * inv;
                if (j < NC) cp[j] = cpj;
                dp[j] = dpj;
                w_prev = w; u_prev = u; cp_prev = cpj; dp_prev = dpj;
            }
            // back substitution: dz overwrites dp
            #pragma unroll
            for (int j = NN-2; j >= 0; --j) dp[j] = dp[j] - cp[j]*dp[j+1];

            // fraction-to-boundary step length
            float a = 1.0f;
            #pragma unroll
            for (int k = 0; k < NC; ++k) {
                float d  = zz[k] - zz[k+1];
                float gd = dp[k] - dp[k+1];
                float da = -(( d + sA[k] - C) + gd);
                float db = -((-d + sB[k] - C) - gd);
                float dla = (muc - lA[k]*(sA[k]+da)) * frcp(sA[k]);
                float dlb = (muc - lB[k]*(sB[k]+db)) * frcp(sB[k]);
                a = fminf(a, (da  < 0.f) ? -sA[k]*frcp(da)  : 1.0f);
                a = fminf(a, (db  < 0.f) ? -sB[k]*frcp(db)  : 1.0f);
                a = fminf(a, (dla < 0.f) ? -lA[k]*frcp(dla) : 1.0f);
                a = fminf(a, (dlb < 0.f) ? -lB[k]*frcp(dlb) : 1.0f);
            }
            a *= 0.99f;

            // update (recompute directions: cheaper than spilling 92 VGPRs)
            #pragma unroll
            for (int k = 0; k < NC; ++k) {
                float d  = zz[k] - zz[k+1];
                float gd = dp[k] - dp[k+1];
                float da = -(( d + sA[k] - C) + gd);
                float db = -((-d + sB[k] - C) - gd);
                float dla = (muc - lA[k]*(sA[k]+da)) * frcp(sA[k]);
                float dlb = (muc - lB[k]*(sB[k]+db)) * frcp(sB[k]);
                sA[k] += a*da;  sB[k] += a*db;
                lA[k] += a*dla; lB[k] += a*dlb;
            }
            #pragma unroll
            for (int j = 0; j < NN; ++j) zz[j] += a*dp[j];
        }
    }

    // ---- write result: 24 contiguous floats per problem, 128b stores
    if (prob < batch) {
        float4* o = (float4*)(gout + (size_t)prob * NN);
        #pragma unroll
        for (int q = 0; q < NN/4; ++q)
            o[q] = make_float4(zz[4*q+0], zz[4*q+1], zz[4*q+2], zz[4*q+3]);
    }
}

extern "C" void kernel_launch(void* const* d_in, const int* in_sizes, int n_in,
                              void* d_out, int out_size, void* d_ws, size_t ws_size,
                              hipStream_t stream) {
    (void)n_in; (void)out_size; (void)d_ws; (void)ws_size;
    const float* mu  = (const float*)d_in[0];
    const float* sig = (const float*)d_in[1];
    float* out = (float*)d_out;
    int batch = in_sizes[0] / NN;
    int grid  = (batch + TPB - 1) / TPB;
    hipLaunchKernelGGL(sqp_ipm_kernel, dim3(grid), dim3(TPB), 0, stream,
                       mu, sig, out, batch);
}